// Encoder1_5437428597182
// MI455X (gfx1250) — compile-verified
//
#include <hip/hip_runtime.h>
#include <hip/hip_bf16.h>
#include <math.h>

// ---- problem constants (match reference) ----
#define BDIM   128          // batch
#define NF     1228
#define CHDIM  1229         // NF + 1
#define HDIM   64
#define H1DIM  128
#define OUT2   (HDIM * CHDIM)   // 78656
#define NSTEPS 6
#define DT_    0.5f         // 3.0 / 6

#define NCHUNK 8            // c-chunks per hh
#define CCHUNK 160          // 8 * 160 = 1280 >= 1229 (padded, masked via dX)
#define NTILES (CCHUNK / 16)

typedef __attribute__((ext_vector_type(2))) float v2f;
typedef __attribute__((ext_vector_type(8))) float v8f;

// ---- CDNA5 async global->LDS via inline asm (ISA 15.18.3 op 96) ----------
// VDST = per-lane LDS byte offset (low 32 bits of a generic __shared__ ptr),
// VADDR = 64-bit global address, SADDR = off. Tracked by ASYNCcnt.
__device__ __forceinline__ void async_ld_b32(unsigned lds_off,
                                             unsigned long long gaddr) {
    asm volatile("global_load_async_to_lds_b32 %0, %1, off"
                 :: "v"(lds_off), "v"(gaddr) : "memory");
}
__device__ __forceinline__ void async_wait0() {
    asm volatile("s_wait_asynccnt 0x0" ::: "memory");
}

// ---------------------------------------------------------------------------
// Precompute dXdt planes for the 13 distinct t values (t = q*0.25, q=0..12).
// Hermite coeffs per interval i from diffs d0,d1,d2:
//   i=0: b=d0, c=0,           d=0
//   i=1: b=d0, c=2(d1-d0),    d=d0-d1
//   i=2: b=d1, c=2(d2-d1),    d=d1-d2
// dXdt = b + 2 c fr + 3 d fr^2
// ---------------------------------------------------------------------------
__global__ void prep_dx_kernel(const float* __restrict__ x, float* __restrict__ dXall) {
    int i = blockIdx.x * blockDim.x + threadIdx.x;
    if (i >= BDIM * CHDIM) return;
    int b = i / CHDIM;
    int c = i % CHDIM;
    float p0, p1, p2, p3;
    if (c == 0) { p0 = 0.0f; p1 = 1.0f; p2 = 2.0f; p3 = 3.0f; }  // time channel
    else {
        const float* xp = x + (size_t)b * NF + (c - 1);          // x[j, b, c-1]
        p0 = xp[0];
        p1 = xp[(size_t)1 * BDIM * NF];
        p2 = xp[(size_t)2 * BDIM * NF];
        p3 = xp[(size_t)3 * BDIM * NF];
    }
    float d0 = p1 - p0, d1 = p2 - p1, d2 = p3 - p2;
    float bb[3] = { d0, d0, d1 };
    float cc[3] = { 0.0f, 2.0f * (d1 - d0), 2.0f * (d2 - d1) };
    float dd[3] = { 0.0f, d0 - d1, d1 - d2 };
    #pragma unroll
    for (int q = 0; q < 13; ++q) {
        int idx = q >> 2; if (idx > 2) idx = 2;
        float fr = 0.25f * (float)q - (float)idx;
        float v = bb[idx] + 2.0f * cc[idx] * fr + 3.0f * dd[idx] * fr * fr;
        dXall[(size_t)q * BDIM * CHDIM + i] = v;
    }
}

// ---------------------------------------------------------------------------
// z0 = inputs[:,0,:] @ Wi + bi   (inputs[:,0,0] = t0 = 0, so channel 0 drops)
// ---------------------------------------------------------------------------
__global__ __launch_bounds__(HDIM)
void z0_kernel(const float* __restrict__ x, const float* __restrict__ Wi,
               const float* __restrict__ bi, float* __restrict__ z) {
    int b = blockIdx.x;
    int hcol = threadIdx.x;
    const float* xrow = x + (size_t)b * NF;   // x[0, b, :]
    float acc = bi[hcol];
    for (int c = 1; c < CHDIM; ++c)
        acc = fmaf(xrow[c - 1], Wi[c * HDIM + hcol], acc);
    z[b * HDIM + hcol] = acc;
}

// ---------------------------------------------------------------------------
// hstage: ztmp = z + alpha*kin;  h = relu(ztmp @ W1 + b1)
// Also zeroes the stage accumulator kzero (vf_kernel accumulates atomically).
// ---------------------------------------------------------------------------
__global__ __launch_bounds__(H1DIM)
void hstage_kernel(const float* __restrict__ z, const float* __restrict__ kin, float alpha,
                   const float* __restrict__ W1, const float* __restrict__ b1,
                   float* __restrict__ h, float* __restrict__ kzero) {
    __shared__ float zrow[HDIM];
    int b = blockIdx.x;
    int j = threadIdx.x;
    int gid = b * H1DIM + j;
    if (gid < BDIM * HDIM) kzero[gid] = 0.0f;
    if (j < HDIM) zrow[j] = z[b * HDIM + j] + alpha * kin[b * HDIM + j];
    __syncthreads();
    float acc = b1[j];
    #pragma unroll 8
    for (int i = 0; i < HDIM; ++i)
        acc = fmaf(zrow[i], W1[i * H1DIM + j], acc);
    h[(size_t)b * H1DIM + j] = fmaxf(acc, 0.0f);
}

// ---------------------------------------------------------------------------
// vf mega-kernel:  kout[b,hh] += sum_c tanh( (h@W2)[b, hh*CH+c] + b2 ) * dX[b,c]
// grid = (64 hh, 8 c-chunks), 256 threads = 8 waves; wave w owns batches
// [16w, 16w+16). W2 tiles staged in LDS in WMMA B-fragment order
// ([ks][lane][2] f32), double-buffered with async global->LDS DMA; all 32
// B-fragments are preloaded to VGPRs so the 32 v_wmma_f32_16x16x4_f32 issue
// back-to-back (two independent accumulator chains).
// ---------------------------------------------------------------------------
__global__ __launch_bounds__(256)
void vf_kernel(const float* __restrict__ h,   // B x H1
               const float* __restrict__ W2,  // H1 x OUT2 (row-major)
               const float* __restrict__ b2,  // OUT2
               const float* __restrict__ dX,  // B x CH  (dXdt at this stage's t)
               float* __restrict__ kout)      // B x H   (pre-zeroed; atomic accum)
{
    __shared__ float ldsB[2][H1DIM * 16];     // 2 x 8 KB, fragment-ordered

    const int hh    = blockIdx.x;             // 0..63 output hidden index
    const int chunk = blockIdx.y;             // 0..7  c-chunk
    const int tid   = threadIdx.x;
    const int wave  = tid >> 5;
    const int lane  = tid & 31;
    const int lhalf = lane >> 4;              // 0 | 1
    const int l16   = lane & 15;

    const int m0 = wave * 16;                 // batch tile base for this wave
    const size_t colBase = (size_t)hh * CHDIM;
    const int cBase = chunk * CCHUNK;

    // ---- per-thread tile-load mapping (8 elements/thread, fragment order) --
    // element e: idx = tid + 256*e ; kk = idx>>4 ; cc = idx&15
    // fragIdx = (kk>>2)*64 + (((kk&3)>>1)*16 + cc)*2 + (kk&1)
    int fkk[8], fcc[8], ffrag[8];
    #pragma unroll
    for (int e = 0; e < 8; ++e) {
        int idx = tid + 256 * e;
        int kk = idx >> 4, cc = idx & 15;
        int ks = kk >> 2, kl = kk & 3;
        fkk[e]   = kk;
        fcc[e]   = cc;
        ffrag[e] = ks * 64 + (((kl >> 1) << 4) + cc) * 2 + (kl & 1);
    }

    // A fragments for all K=128: 32 k-steps, v2f each.
    // ISA layout (f32 16x4 A): lanes 0-15 -> K={0,1}, lanes 16-31 -> K={2,3}.
    v2f aFrag[32];
    {
        const float* hrow = h + (size_t)(m0 + l16) * H1DIM;
        #pragma unroll
        for (int ks = 0; ks < 32; ++ks) {
            int k = ks * 4 + 2 * lhalf;
            aFrag[ks].x = hrow[k];
            aFrag[ks].y = hrow[k + 1];
        }
    }

    float acc[8];
    #pragma unroll
    for (int r = 0; r < 8; ++r) acc[r] = 0.0f;

    // ---- prologue: async-load tile 0 into buffer 0 ----
    {
        #pragma unroll
        for (int e = 0; e < 8; ++e) {
            int c = cBase + fcc[e];
            int ccl = (c < CHDIM) ? c : (CHDIM - 1);
            async_ld_b32((unsigned)(size_t)&ldsB[0][ffrag[e]],
                         (unsigned long long)(size_t)(W2 + (size_t)fkk[e] * OUT2 + colBase + ccl));
        }
    }

    for (int nt = 0; nt < NTILES; ++nt) {
        const int c0 = cBase + nt * 16;
        const int buf = nt & 1;

        async_wait0();            // this wave's loads for tile nt have landed
        __syncthreads();          // everyone's landed; nobody reads buf^1 anymore
        if (nt + 1 < NTILES) {    // prefetch tile nt+1 into the other buffer
            const int c1 = c0 + 16;
            #pragma unroll
            for (int e = 0; e < 8; ++e) {
                int c = c1 + fcc[e];
                int ccl = (c < CHDIM) ? c : (CHDIM - 1);
                async_ld_b32((unsigned)(size_t)&ldsB[buf ^ 1][ffrag[e]],
                             (unsigned long long)(size_t)(W2 + (size_t)fkk[e] * OUT2 + colBase + ccl));
            }
        }

        // ---- preload all B fragments (conflict-free b64 per lane) ----
        v2f bFrag[32];
        {
            const v2f* bp = (const v2f*)(&ldsB[buf][0]);
            #pragma unroll
            for (int ks = 0; ks < 32; ++ks)
                bFrag[ks] = bp[ks * 32 + lane];
        }

        // ---- 32 back-to-back WMMAs, two independent accumulator chains ----
        v8f D0 = {}, D1 = {};
        #pragma unroll
        for (int ks = 0; ks < 32; ks += 2) {
            D0 = __builtin_amdgcn_wmma_f32_16x16x4_f32(
                     false, aFrag[ks],     false, bFrag[ks],     (short)0, D0, false, false);
            D1 = __builtin_amdgcn_wmma_f32_16x16x4_f32(
                     false, aFrag[ks + 1], false, bFrag[ks + 1], (short)0, D1, false, false);
        }

        // Epilogue in D layout: VGPR r -> (M = m0 + r + 8*lhalf, N = l16).
        int c = c0 + l16;
        int ccl  = (c < CHDIM) ? c : (CHDIM - 1);
        float msk = (c < CHDIM) ? 1.0f : 0.0f;
        float bias = b2[colBase + ccl];
        #pragma unroll
        for (int r = 0; r < 8; ++r) {
            int b = m0 + r + 8 * lhalf;
            float dx = dX[(size_t)b * CHDIM + ccl] * msk;
            acc[r] += tanhf((D0[r] + D1[r]) + bias) * dx;
        }
    }

    // Reduce over the 16 N-lanes within each half-wave, then accumulate.
    #pragma unroll
    for (int r = 0; r < 8; ++r) {
        float v = acc[r];
        #pragma unroll
        for (int off = 8; off >= 1; off >>= 1)
            v += __shfl_xor(v, off, 16);
        if (l16 == 0) {
            int b = m0 + r + 8 * lhalf;
            atomicAdd(&kout[b * HDIM + hh], v);
        }
    }
}

// ---------------------------------------------------------------------------
// RK4 combine: z += (DT/6) * (k1 + 2 k2 + 2 k3 + k4)
// ---------------------------------------------------------------------------
__global__ void combine_kernel(float* __restrict__ z,
                               const float* __restrict__ k1, const float* __restrict__ k2,
                               const float* __restrict__ k3, const float* __restrict__ k4) {
    int i = blockIdx.x * blockDim.x + threadIdx.x;
    if (i < BDIM * HDIM)
        z[i] += (DT_ / 6.0f) * (k1[i] + 2.0f * k2[i] + 2.0f * k3[i] + k4[i]);
}

__global__ void finalize_kernel(const float* __restrict__ z, float* __restrict__ out) {
    int i = blockIdx.x * blockDim.x + threadIdx.x;
    if (i < BDIM * HDIM) {
        out[i] = z[i];                 // hn
        out[BDIM * HDIM + i] = 0.0f;   // cn = zeros
    }
}

// ---------------------------------------------------------------------------
extern "C" void kernel_launch(void* const* d_in, const int* in_sizes, int n_in,
                              void* d_out, int out_size, void* d_ws, size_t ws_size,
                              hipStream_t stream) {
    const float* x  = (const float*)d_in[0];
    const float* Wi = (const float*)d_in[1];
    const float* bi = (const float*)d_in[2];
    const float* W1 = (const float*)d_in[3];
    const float* b1 = (const float*)d_in[4];
    const float* W2 = (const float*)d_in[5];
    const float* b2 = (const float*)d_in[6];
    float* out = (float*)d_out;

    float* ws    = (float*)d_ws;
    float* dXall = ws;                                   // 13 * B * CH
    float* z     = dXall + (size_t)13 * BDIM * CHDIM;    // B * H
    float* h     = z + BDIM * HDIM;                      // B * H1
    float* kbuf  = h + BDIM * H1DIM;                     // 4 * B * H

    prep_dx_kernel<<<(BDIM * CHDIM + 255) / 256, 256, 0, stream>>>(x, dXall);
    z0_kernel<<<BDIM, HDIM, 0, stream>>>(x, Wi, bi, z);

    for (int n = 0; n < NSTEPS; ++n) {
        for (int s = 0; s < 4; ++s) {
            float alpha = (s == 0) ? 0.0f : (s == 3) ? 0.5f : 0.25f;
            const float* kin = (s == 0) ? z : (kbuf + (size_t)(s - 1) * BDIM * HDIM);
            float* ko = kbuf + (size_t)s * BDIM * HDIM;
            int q = (s == 0) ? 2 * n : (s == 3) ? (2 * n + 2) : (2 * n + 1);
            hstage_kernel<<<BDIM, H1DIM, 0, stream>>>(z, kin, alpha, W1, b1, h, ko);
            vf_kernel<<<dim3(HDIM, NCHUNK), 256, 0, stream>>>(
                h, W2, b2, dXall + (size_t)q * BDIM * CHDIM, ko);
        }
        combine_kernel<<<(BDIM * HDIM + 255) / 256, 256, 0, stream>>>(
            z, kbuf, kbuf + BDIM * HDIM, kbuf + 2 * BDIM * HDIM, kbuf + 3 * BDIM * HDIM);
    }
    finalize_kernel<<<(BDIM * HDIM + 255) / 256, 256, 0, stream>>>(z, out);
}